// ModelNew_36215164240202
// MI455X (gfx1250) — compile-verified
//
#include <hip/hip_runtime.h>
#include <hip/hip_bf16.h>

typedef __attribute__((ext_vector_type(2))) float v2f;
typedef __attribute__((ext_vector_type(8))) float v8f;

#define WMMA_F32(a, b, c) \
    __builtin_amdgcn_wmma_f32_16x16x4_f32(false, (a), false, (b), (short)0, (c), false, false)

// Constants for this problem instance
#define GRU_T 512
#define GRU_B 64
#define GRU_H 256
#define GRU_M (GRU_T * GRU_B)   // 32768 rows
#define GRU_N 768               // 3H

// ---------------------------------------------------------------------------
// Weight prep: materialize W_eff^T in [K][768] layout.
// Reference does _eff(W) = W.T.reshape(W.shape) (buffer reinterpretation):
//   W_eff[g,k] = Wt_flat[g*in + k], where Wt_flat[a*768 + b] = W[b*in + a].
// We store WT[k*768 + g] = W_eff[g,k] so GEMMs read B-fragments row-major in K.
// ---------------------------------------------------------------------------
__global__ void gru_prep_wt(const float* __restrict__ W,  // [2][768][in]
                            float* __restrict__ WT,        // [2][in][768]
                            int in_dim) {
    const int d = blockIdx.y;
    const int total = GRU_N * in_dim;
    int f = blockIdx.x * 256 + threadIdx.x;
    if (f >= total) return;
    int a = f / GRU_N;            // decode as W.T flat index
    int b = f % GRU_N;
    float v = W[(size_t)d * total + (size_t)b * in_dim + a];
    int g = f / in_dim;           // decode as [g,k] of W_eff
    int k = f % in_dim;
    WT[(size_t)d * total + (size_t)k * GRU_N + g] = v;
}

// ---------------------------------------------------------------------------
// Input projection: gi[d][m][g] = sum_k X[m][k] * WT[d][k][g] + b_ih[d][g]
// Grid: (M/16, 2 dirs), 256 threads (8 waves), wave w -> n-tiles [6w, 6w+6)
// ---------------------------------------------------------------------------
__global__ void __launch_bounds__(256)
gru_gi_gemm(const float* __restrict__ X,    // [M][K]
            const float* __restrict__ WT,   // [2][K][768]
            const float* __restrict__ bih,  // [2][768]
            float* __restrict__ gi,         // [2][M][768]
            int K) {
    const int mt   = blockIdx.x;
    const int d    = blockIdx.y;
    const int wv   = threadIdx.x >> 5;
    const int lane = threadIdx.x & 31;
    const int l16  = lane & 15;
    const int lh   = lane >> 4;

    const float* wt   = WT + (size_t)d * K * GRU_N;
    const float* xrow = X + (size_t)(mt * 16 + l16) * K;  // A row for this lane
    const int nt0 = wv * 6;

    v8f acc0 = {}, acc1 = {}, acc2 = {}, acc3 = {}, acc4 = {}, acc5 = {};

    for (int k4 = 0; k4 < K; k4 += 4) {
        const int ka = k4 + 2 * lh;
        v2f a;
        a.x = xrow[ka];
        a.y = xrow[ka + 1];
        const float* wrow = wt + (size_t)ka * GRU_N;
        v2f b;
        int n;
        n = (nt0 + 0) * 16 + l16; b.x = wrow[n]; b.y = wrow[GRU_N + n]; acc0 = WMMA_F32(a, b, acc0);
        n = (nt0 + 1) * 16 + l16; b.x = wrow[n]; b.y = wrow[GRU_N + n]; acc1 = WMMA_F32(a, b, acc1);
        n = (nt0 + 2) * 16 + l16; b.x = wrow[n]; b.y = wrow[GRU_N + n]; acc2 = WMMA_F32(a, b, acc2);
        n = (nt0 + 3) * 16 + l16; b.x = wrow[n]; b.y = wrow[GRU_N + n]; acc3 = WMMA_F32(a, b, acc3);
        n = (nt0 + 4) * 16 + l16; b.x = wrow[n]; b.y = wrow[GRU_N + n]; acc4 = WMMA_F32(a, b, acc4);
        n = (nt0 + 5) * 16 + l16; b.x = wrow[n]; b.y = wrow[GRU_N + n]; acc5 = WMMA_F32(a, b, acc5);
    }

    float* gid = gi + (size_t)d * GRU_M * GRU_N;
    v8f accs[6] = {acc0, acc1, acc2, acc3, acc4, acc5};
#pragma unroll
    for (int i = 0; i < 6; ++i) {
        const int n = (nt0 + i) * 16 + l16;
        const float bias = bih[d * GRU_N + n];
#pragma unroll
        for (int r = 0; r < 8; ++r) {
            const int m = mt * 16 + 8 * lh + r;  // C/D layout: M = vgpr + 8*(lane>=16)
            gid[(size_t)m * GRU_N + n] = accs[i][r] + bias;
        }
    }
}

// ---------------------------------------------------------------------------
// Recurrent scan. Grid: (4 batch tiles, 2 dirs), 512 threads = 16 waves.
// Each block owns 16 batch rows; hidden slice [16][256] double-buffered in LDS.
// Wave w handles j-tile w: three accumulators (r, z, n gates) whose C-layout
// coordinates (b, j) match, so gates + h-update are register-local.
// ---------------------------------------------------------------------------
__global__ void __launch_bounds__(512)
gru_scan(const float* __restrict__ gi,   // [2][M][768]
         const float* __restrict__ WHT,  // [2][256][768]
         const float* __restrict__ bhh,  // [2][768]
         const float* __restrict__ h0,   // [2][64][256]  (this layer's slice)
         float* __restrict__ out) {      // [T][64][512]
    __shared__ float hbuf[2][16 * GRU_H];

    const int mt   = blockIdx.x;   // batch tile (16 rows)
    const int d    = blockIdx.y;   // direction
    const int wv   = threadIdx.x >> 5;  // == j-tile, 0..15
    const int lane = threadIdx.x & 31;
    const int l16  = lane & 15;
    const int lh   = lane >> 4;

    // Load initial hidden state slice into LDS
    const float* h0p = h0 + (size_t)d * GRU_B * GRU_H + (size_t)mt * 16 * GRU_H;
    for (int i = threadIdx.x; i < 16 * GRU_H; i += blockDim.x)
        hbuf[0][i] = h0p[i];
    __syncthreads();

    const float* wt  = WHT + (size_t)d * GRU_H * GRU_N;
    const float* bh  = bhh + d * GRU_N;
    const float* gid = gi + (size_t)d * GRU_M * GRU_N;

    const int j   = wv * 16 + l16;     // hidden index this lane produces
    const float bhr = bh[j];
    const float bhz = bh[GRU_H + j];
    const float bhn = bh[2 * GRU_H + j];

    int cur = 0;
    for (int step = 0; step < GRU_T; ++step) {
        const int t = d ? (GRU_T - 1 - step) : step;
        const float* h = hbuf[cur];

        v8f cr = {}, cz = {}, cn = {};
        for (int k4 = 0; k4 < GRU_H; k4 += 4) {
            const int ka = k4 + 2 * lh;
            v2f a;
            a.x = h[l16 * GRU_H + ka];       // A: m = lane&15, k = 2*(lane>=16)+vgpr
            a.y = h[l16 * GRU_H + ka + 1];
            const float* wrow = wt + (size_t)ka * GRU_N;
            const int nr = wv * 16 + l16;
            v2f b;
            b.x = wrow[nr];                   b.y = wrow[GRU_N + nr];                   cr = WMMA_F32(a, b, cr);
            b.x = wrow[GRU_H + nr];           b.y = wrow[GRU_N + GRU_H + nr];           cz = WMMA_F32(a, b, cz);
            b.x = wrow[2 * GRU_H + nr];       b.y = wrow[GRU_N + 2 * GRU_H + nr];       cn = WMMA_F32(a, b, cn);
        }

        float* hn = hbuf[cur ^ 1];
#pragma unroll
        for (int r = 0; r < 8; ++r) {
            const int brow_loc = 8 * lh + r;          // local batch row (C-layout M)
            const int brow = mt * 16 + brow_loc;      // global batch row
            const size_t row = (size_t)t * GRU_B + brow;
            const float gir = gid[row * GRU_N + j];
            const float giz = gid[row * GRU_N + GRU_H + j];
            const float gin = gid[row * GRU_N + 2 * GRU_H + j];
            const float rg = 1.0f / (1.0f + __expf(-(gir + cr[r] + bhr)));
            const float zg = 1.0f / (1.0f + __expf(-(giz + cz[r] + bhz)));
            const float ng = tanhf(gin + rg * (cn[r] + bhn));
            const float hold = h[brow_loc * GRU_H + j];
            const float hv = (1.0f - zg) * ng + zg * hold;
            hn[brow_loc * GRU_H + j] = hv;
            out[row * (2 * GRU_H) + d * GRU_H + j] = hv;
        }
        __syncthreads();
        cur ^= 1;
    }
}

// ---------------------------------------------------------------------------
extern "C" void kernel_launch(void* const* d_in, const int* in_sizes, int n_in,
                              void* d_out, int out_size, void* d_ws, size_t ws_size,
                              hipStream_t stream) {
    const float* x      = (const float*)d_in[0];
    const float* h0     = (const float*)d_in[1];
    const float* w_ih_0 = (const float*)d_in[2];
    const float* w_hh_0 = (const float*)d_in[3];
    const float* b_ih_0 = (const float*)d_in[4];
    const float* b_hh_0 = (const float*)d_in[5];
    const float* w_ih_r = (const float*)d_in[6];
    const float* w_hh_r = (const float*)d_in[7];
    const float* b_ih_r = (const float*)d_in[8];
    const float* b_hh_r = (const float*)d_in[9];
    float* outp = (float*)d_out;

    float* ws = (float*)d_ws;
    float* gi   = ws;  ws += (size_t)2 * GRU_M * GRU_N;   // 50.3M floats
    float* actA = ws;  ws += (size_t)GRU_M * 512;         // 16.8M floats
    float* actB = ws;  ws += (size_t)GRU_M * 512;
    float* wihT = ws;  ws += (size_t)2 * 512 * GRU_N;
    float* whhT = ws;  ws += (size_t)2 * GRU_H * GRU_N;

    for (int l = 0; l < 6; ++l) {
        const int in_dim = (l == 0) ? 128 : 512;
        const float* Wih = (l == 0) ? w_ih_0 : (w_ih_r + (size_t)(l - 1) * 2 * GRU_N * 512);
        const float* Whh = (l == 0) ? w_hh_0 : (w_hh_r + (size_t)(l - 1) * 2 * GRU_N * GRU_H);
        const float* bih = (l == 0) ? b_ih_0 : (b_ih_r + (size_t)(l - 1) * 2 * GRU_N);
        const float* bhh = (l == 0) ? b_hh_0 : (b_hh_r + (size_t)(l - 1) * 2 * GRU_N);
        const float* X = (l == 0) ? x : ((l & 1) ? actA : actB);
        float* Y = (l == 5) ? outp : ((l & 1) ? actB : actA);

        {
            const int total = GRU_N * in_dim;
            dim3 g1((total + 255) / 256, 2);
            gru_prep_wt<<<g1, 256, 0, stream>>>(Wih, wihT, in_dim);
            dim3 g2((GRU_N * GRU_H + 255) / 256, 2);
            gru_prep_wt<<<g2, 256, 0, stream>>>(Whh, whhT, GRU_H);
        }

        gru_gi_gemm<<<dim3(GRU_M / 16, 2), 256, 0, stream>>>(X, wihT, bih, gi, in_dim);

        gru_scan<<<dim3(4, 2), 512, 0, stream>>>(
            gi, whhT, bhh, h0 + (size_t)l * 2 * GRU_B * GRU_H, Y);
    }
}